// TSKDLoss_25374666784939
// MI455X (gfx1250) — compile-verified
//
#include <hip/hip_runtime.h>

// ---- problem constants (from reference setup_inputs) ----
#define N_IMG  4
#define C_CH   256
#define H_F    136
#define W_F    200
#define HW     (H_F * W_F)        // 27200
#define CHW    (C_CH * HW)        // 6963200
#define NT     20
#define NP     (N_IMG * HW)       // 108800 pixels total
#define NQ     (N_IMG * CHW / 4)  // 6963200 float4s over whole tensor
#define HW4    (HW / 4)           // 6800
#define CHW4   (CHW / 4)          // 1740800
#define K6_THREADS 256
#define K6_BLOCKS  3400           // K6_BLOCKS*K6_THREADS*K6_ITERS == NQ exactly (no tail)
#define K6_ITERS   8
#define K6_TOT     (K6_BLOCKS * K6_THREADS)   // 870400
#define PARTS      K6_BLOCKS
#define PARTS_PAD  3456           // next multiple of 64 for the WMMA reducer
#define SCALE  (0.0005f / 4.0f)   // ALPHA / N  (fg and bg alphas equal)

typedef __attribute__((ext_vector_type(4))) float f4;
typedef __attribute__((ext_vector_type(2))) float v2f;
typedef __attribute__((ext_vector_type(8))) float v8f;
typedef __attribute__((ext_vector_type(4))) int   v4i;

#if defined(__gfx1250__) && __has_builtin(__builtin_amdgcn_global_load_async_to_lds_b128) && __has_builtin(__builtin_amdgcn_s_wait_asynccnt)
#define ASYNC_OK 1
// builtin signature (from hipcc diagnostic): param0 = AS1 (global) v4i*, 4 args total
#define ASYNC_COPY_B128(gsrc, ldst)                                            \
    __builtin_amdgcn_global_load_async_to_lds_b128(                            \
        (__attribute__((address_space(1))) v4i*)(gsrc),                        \
        (__attribute__((address_space(3))) v4i*)(ldst), 0, 0)
#else
#define ASYNC_OK 0
#endif

#if __has_builtin(__builtin_amdgcn_sched_barrier)
#define SCHED_FENCE() __builtin_amdgcn_sched_barrier(0)
#else
#define SCHED_FENCE() asm volatile("" ::: "memory")
#endif

// ---- K0: zero bg-pixel counters + zero the WMMA-reducer padding slots ----
__global__ void k_init(int* __restrict__ bgc, float* __restrict__ part) {
    if (threadIdx.x < N_IMG) bgc[threadIdx.x] = 0;
    int i = PARTS + (int)threadIdx.x;
    if (i < PARTS_PAD) part[i] = 0.0f;
}

// ---- K1: fea_sum[n,h,w] = sum_c |T|  (pulls preds_T into the 192MB L2) ----
__global__ void k_fea_sum(const float* __restrict__ T, float* __restrict__ fea) {
    int p   = blockIdx.x * blockDim.x + threadIdx.x;   // exact NP threads
    int n   = p / HW;
    int phw = p - n * HW;
    const float* base = T + (size_t)n * CHW + phw;
    float s = 0.0f;
#pragma unroll 8
    for (int c = 0; c < C_CH; ++c) s += fabsf(base[(size_t)c * HW]);
    fea[p] = s;
}

// ---- K2: chan_sum[n,c] = sum_hw |T|  (T served from L2) ----
__global__ void k_chan_sum(const float* __restrict__ T, float* __restrict__ chan) {
    __shared__ float sd[256];
    const f4* base = (const f4*)(T + (size_t)blockIdx.x * HW);
    float s = 0.0f;
    for (int i = threadIdx.x; i < HW4; i += 256) {
        f4 v = base[i];
        s += fabsf(v.x) + fabsf(v.y) + fabsf(v.z) + fabsf(v.w);
    }
    sd[threadIdx.x] = s; __syncthreads();
    for (int st = 128; st > 0; st >>= 1) {
        if ((int)threadIdx.x < st) sd[threadIdx.x] += sd[threadIdx.x + st];
        __syncthreads();
    }
    if (threadIdx.x == 0) chan[blockIdx.x] = sd[0];
}

// ---- K3: S_att[n,h,w] = HW * softmax(fea_mean / TEMP) ; one block per image ----
__global__ void k_satt(const float* __restrict__ fea, float* __restrict__ satt) {
    __shared__ float sd[1024];
    const int n = blockIdx.x;
    const float* f = fea + n * HW;
    float* o = satt + n * HW;
    const float sc = 1.0f / 128.0f;  // (1/C) / TEMP = (1/256)/0.5
    float mx = -3.402823466e38f;
    for (int i = threadIdx.x; i < HW; i += 1024) mx = fmaxf(mx, f[i] * sc);
    sd[threadIdx.x] = mx; __syncthreads();
    for (int st = 512; st > 0; st >>= 1) {
        if ((int)threadIdx.x < st) sd[threadIdx.x] = fmaxf(sd[threadIdx.x], sd[threadIdx.x + st]);
        __syncthreads();
    }
    mx = sd[0]; __syncthreads();
    float se = 0.0f;
    for (int i = threadIdx.x; i < HW; i += 1024) se += expf(f[i] * sc - mx);
    sd[threadIdx.x] = se; __syncthreads();
    for (int st = 512; st > 0; st >>= 1) {
        if ((int)threadIdx.x < st) sd[threadIdx.x] += sd[threadIdx.x + st];
        __syncthreads();
    }
    const float inv = (float)HW / sd[0];
    for (int i = threadIdx.x; i < HW; i += 1024) o[i] = expf(f[i] * sc - mx) * inv;
}

// ---- K4: C_att[n,c] = C * softmax(chan_mean / TEMP) ; one block per image ----
__global__ void k_catt(const float* __restrict__ chan, float* __restrict__ catt) {
    __shared__ float sd[256];
    const int n = blockIdx.x;
    const float sc = 1.0f / 13600.0f;  // (1/HW)/TEMP
    float v = chan[n * C_CH + threadIdx.x] * sc;
    sd[threadIdx.x] = v; __syncthreads();
    for (int st = 128; st > 0; st >>= 1) {
        if ((int)threadIdx.x < st) sd[threadIdx.x] = fmaxf(sd[threadIdx.x], sd[threadIdx.x + st]);
        __syncthreads();
    }
    float mx = sd[0]; __syncthreads();
    float e = expf(v - mx);
    sd[threadIdx.x] = e; __syncthreads();
    for (int st = 128; st > 0; st >>= 1) {
        if ((int)threadIdx.x < st) sd[threadIdx.x] += sd[threadIdx.x + st];
        __syncthreads();
    }
    catt[n * C_CH + threadIdx.x] = e * ((float)C_CH / sd[0]);
}

// ---- K5: Mask_fg scatter-max + bg pixel counts (integer atomics = deterministic) ----
__global__ void k_mask(const float* __restrict__ gt, float* __restrict__ mfg,
                       int* __restrict__ bgc) {
    __shared__ float bx[N_IMG * NT * 4];
    for (int i = threadIdx.x; i < N_IMG * NT * 4; i += blockDim.x) bx[i] = gt[i];
    __syncthreads();
    int p   = blockIdx.x * blockDim.x + threadIdx.x;
    int n   = p / HW;
    int phw = p - n * HW;
    int hi  = phw / W_F;
    float h = (float)hi;
    float w = (float)(phw - hi * W_F);
    const float* B = bx + n * NT * 4;
    float m = 0.0f;
    for (int t = 0; t < NT; ++t) {
        float x1 = B[t * 4 + 0], y1 = B[t * 4 + 1];
        float x2 = B[t * 4 + 2], y2 = B[t * 4 + 3];
        float wmin = floorf(x1 / 1600.0f * 200.0f);
        float wmax = ceilf (x2 / 1600.0f * 200.0f);
        float hmin = floorf(y1 / 1088.0f * 136.0f);
        float hmax = ceilf (y2 / 1088.0f * 136.0f);
        float area = 1.0f / ((hmax + 1.0f - hmin) * (wmax + 1.0f - wmin));
        bool in = (h >= hmin) && (h <= hmax) && (w >= wmin) && (w <= wmax);
        m = fmaxf(m, in ? area : 0.0f);
    }
    mfg[p] = m;
    if (m == 0.0f) atomicAdd(&bgc[n], 1);
}

// ---- K6: fused loss pass.
//   preds_T: double-buffered GLOBAL_LOAD_ASYNC_TO_LDS_B128 (ASYNCcnt, L2 hits),
//            each wave reads back only its own LDS slots -> no barrier needed.
//   preds_S: non-temporal VGPR loads (TH=NT) so the 111MB T stays L2-resident.
//   Uniform 8-iteration schedule: issue/wait counts identical across the wave. ----
__global__ void k_main(const float* __restrict__ S, const float* __restrict__ Tm,
                       const float* __restrict__ satt, const float* __restrict__ mfg,
                       const float* __restrict__ catt, const int* __restrict__ bgc,
                       float* __restrict__ partials) {
    __shared__ float sd[K6_THREADS];
    __shared__ float invbg[N_IMG];
    if (threadIdx.x < N_IMG) {
        int c = bgc[threadIdx.x];
        invbg[threadIdx.x] = (c > 0) ? (1.0f / (float)c) : 0.0f;
    }
    __syncthreads();
    const f4* S4 = (const f4*)S;
    const f4* T4 = (const f4*)Tm;
    const f4* A4 = (const f4*)satt;
    const f4* M4 = (const f4*)mfg;
    const unsigned base = blockIdx.x * K6_THREADS + threadIdx.x;
    float acc = 0.0f;
#if ASYNC_OK
    __shared__ f4 tbuf[2][K6_THREADS];
    ASYNC_COPY_B128(T4 + base, &tbuf[0][threadIdx.x]);
#endif
#pragma unroll
    for (int i = 0; i < K6_ITERS; ++i) {
        unsigned q  = base + (unsigned)i * K6_TOT;
        unsigned n  = q / CHW4;
        unsigned r  = q - n * CHW4;
        unsigned c  = r / HW4;
        unsigned p4 = r - c * HW4;
        unsigned px = n * HW4 + p4;
        f4 s = __builtin_nontemporal_load(&S4[q]);  // TH=NT: don't evict T from L2
        f4 a = A4[px];
        f4 m = M4[px];
        float ib = invbg[n];
        float ca = catt[n * C_CH + c];
#if ASYNC_OK
        if (i + 1 < K6_ITERS) {
            ASYNC_COPY_B128(T4 + q + K6_TOT, &tbuf[(i + 1) & 1][threadIdx.x]);
            __builtin_amdgcn_s_wait_asynccnt(1);   // oldest tile landed (in-order)
        } else {
            __builtin_amdgcn_s_wait_asynccnt(0);   // drain
        }
        SCHED_FENCE();
        f4 t = tbuf[i & 1][threadIdx.x];
#else
        f4 t = T4[q];
#endif
        float d, ws = 0.0f;
        d = s.x - t.x; ws += d * d * (a.x * (m.x + ((m.x > 0.0f) ? 0.0f : ib)));
        d = s.y - t.y; ws += d * d * (a.y * (m.y + ((m.y > 0.0f) ? 0.0f : ib)));
        d = s.z - t.z; ws += d * d * (a.z * (m.z + ((m.z > 0.0f) ? 0.0f : ib)));
        d = s.w - t.w; ws += d * d * (a.w * (m.w + ((m.w > 0.0f) ? 0.0f : ib)));
        acc += ca * ws;
    }
    sd[threadIdx.x] = acc; __syncthreads();
    for (int st = K6_THREADS / 2; st > 0; st >>= 1) {
        if ((int)threadIdx.x < st) sd[threadIdx.x] += sd[threadIdx.x + st];
        __syncthreads();
    }
    if (threadIdx.x == 0) partials[blockIdx.x] = sd[0];
}

// ---- K7: single wave32 sums PARTS_PAD partials with V_WMMA_F32_16X16X4_F32.
//          A == all-ones  =>  D rows = column sums of B, row-replicated 16x,
//          so sum(all D)/16 == sum(all B) regardless of the B VGPR layout. ----
__global__ void k_final(const float* __restrict__ partials, float* __restrict__ out) {
    const int lane = threadIdx.x;  // exactly 32 threads, EXEC all ones (WMMA requirement)
#if __has_builtin(__builtin_amdgcn_wmma_f32_16x16x4_f32)
    v8f acc = (v8f){0.f, 0.f, 0.f, 0.f, 0.f, 0.f, 0.f, 0.f};
    v2f a;  a.x = 1.0f; a.y = 1.0f;
    for (int i = 0; i < PARTS_PAD; i += 64) {
        v2f b;
        b.x = partials[i + lane];
        b.y = partials[i + 32 + lane];
        acc = __builtin_amdgcn_wmma_f32_16x16x4_f32(false, a, false, b,
                                                    (short)0, acc, false, false);
    }
    float s = acc[0] + acc[1] + acc[2] + acc[3] + acc[4] + acc[5] + acc[6] + acc[7];
    s *= (1.0f / 16.0f);
#else
    float s = 0.0f;
    for (int i = lane; i < PARTS_PAD; i += 32) s += partials[i];
#endif
    for (int off = 16; off > 0; off >>= 1) s += __shfl_xor(s, off, 32);
    if (lane == 0) out[0] = s * SCALE;
}

extern "C" void kernel_launch(void* const* d_in, const int* in_sizes, int n_in,
                              void* d_out, int out_size, void* d_ws, size_t ws_size,
                              hipStream_t stream) {
    (void)in_sizes; (void)n_in; (void)out_size; (void)ws_size;
    const float* S  = (const float*)d_in[0];   // preds_S [4,256,136,200] f32
    const float* T  = (const float*)d_in[1];   // preds_T [4,256,136,200] f32
    const float* gt = (const float*)d_in[2];   // gt_bboxes [4,20,4] f32
    float* out = (float*)d_out;                // scalar f32 loss

    char* w = (char*)d_ws;                     // all offsets 16B-aligned for f4 views
    float* fea  = (float*)w; w += (size_t)NP * 4;
    float* satt = (float*)w; w += (size_t)NP * 4;
    float* mfgv = (float*)w; w += (size_t)NP * 4;
    float* chan = (float*)w; w += (size_t)N_IMG * C_CH * 4;
    float* catt = (float*)w; w += (size_t)N_IMG * C_CH * 4;
    float* part = (float*)w; w += (size_t)PARTS_PAD * 4;
    int*   bgc  = (int*)w;   w += 16;

    k_init    <<<1, 64, 0, stream>>>(bgc, part);
    k_fea_sum <<<NP / 256, 256, 0, stream>>>(T, fea);           // 425 blocks, exact
    k_chan_sum<<<N_IMG * C_CH, 256, 0, stream>>>(T, chan);      // 1024 blocks
    k_satt    <<<N_IMG, 1024, 0, stream>>>(fea, satt);
    k_catt    <<<N_IMG, 256, 0, stream>>>(chan, catt);
    k_mask    <<<NP / 256, 256, 0, stream>>>(gt, mfgv, bgc);
    k_main    <<<K6_BLOCKS, K6_THREADS, 0, stream>>>(S, T, satt, mfgv, catt, bgc, part);
    k_final   <<<1, 32, 0, stream>>>(part, out);
}